// GATNet_54382875902526
// MI455X (gfx1250) — compile-verified
//
#include <hip/hip_runtime.h>

// ---------------------------------------------------------------------------
// 2-layer GAT forward for MI455X (gfx1250, wave32).
// GEMMs: V_WMMA_F32_16X16X4_F32, workgroup-tiled (128x16 per block, 8 waves),
//        B panel staged in LDS via global_load_async_to_lds_b32 (ASYNCcnt),
//        branchless K-tail via clamp+mask, K known at compile time.
// Edge softmax: float atomic-max (bit trick) + global_atomic_add_f32 scatter.
// ---------------------------------------------------------------------------

typedef __attribute__((ext_vector_type(2))) float v2f;
typedef __attribute__((ext_vector_type(8))) float v8f;

#define BLOCK 256
#define USE_ASYNC_LDS 1

constexpr int NN   = 20000;            // nodes
constexpr int E0C  = 256000;           // raw edges
constexpr int ETOT = E0C + NN;         // + self loops
constexpr int NG   = 512;              // graphs
constexpr int FIN  = 75;               // layer-1 per-head feature dim
constexpr int NH   = 10;               // heads
constexpr int HF   = NH * FIN;         // 750
constexpr int D2   = 128;              // layer-2 dim

// ---------------- device helpers ----------------

__device__ __forceinline__ float atomicMaxF(float* addr, float v) {
  if (v >= 0.0f)
    return __int_as_float(atomicMax((int*)addr, __float_as_int(v)));
  else
    return __uint_as_float(atomicMin((unsigned int*)addr, __float_as_uint(v)));
}

__device__ __forceinline__ void edge_sd(const int* __restrict__ ei, int e,
                                        int& s, int& d) {
  if (e < E0C) { s = ei[e]; d = ei[E0C + e]; }
  else         { s = e - E0C; d = s; }
}

// ---------------- fill ----------------

__global__ void gat_fill(unsigned int* __restrict__ p, unsigned int v, long n) {
  long i = (long)blockIdx.x * blockDim.x + threadIdx.x;
  long stride = (long)gridDim.x * blockDim.x;
  for (; i < n; i += stride) p[i] = v;
}

// ---------------- fp32 WMMA GEMM: C = A[MxK] * B[KxN] (+bias, relu) --------
// Block = 8 waves = 128x16 output tile. B panel (K x 16) staged in LDS once
// per block (async global->LDS copies), pair-interleaved so each wave reads
// its two B values as one ds_load_b64. Each wave: one 16x16 WMMA tile chain.
// M must be a multiple of 16 (true for 20000 / 512). N may be ragged.

template <int KK>
__global__ __launch_bounds__(BLOCK) void gat_wmma_gemm(
    const float* __restrict__ A, const float* __restrict__ B,
    float* __restrict__ C, int M, int N,
    const float* __restrict__ bias, int do_relu) {
  constexpr int KPAD  = (KK + 3) & ~3;
  constexpr int KMAIN = KK & ~3;
  __shared__ float lds_b[KPAD * 16];   // element (k,col) at ((k>>1)*16+col)*2+(k&1)

  const int t    = threadIdx.x;
  const int lane = t & 31;
  const int wv   = t >> 5;

  const int ntiles = (N + 15) >> 4;
  const int bm = blockIdx.x / ntiles;
  const int tn = blockIdx.x - bm * ntiles;
  const int n0 = tn << 4;

  // ---- cooperative B panel fill (async to LDS; zeros for OOB) ----
  {
    const int col = t & 15;
    const int nc  = n0 + col;
    for (int k = t >> 4; k < KPAD; k += 16) {
      const unsigned off = (unsigned)(((((k >> 1) * 16) + col) * 2 + (k & 1)) * 4);
      if (k < KK && nc < N) {
#if USE_ASYNC_LDS
        const float* src = B + (long)k * N + nc;
        unsigned dst = (unsigned)(size_t)((char*)lds_b + off);
        asm volatile("global_load_async_to_lds_b32 %0, %1, off"
                     :: "v"(dst), "v"(src) : "memory");
#else
        *(float*)((char*)lds_b + off) = B[(long)k * N + nc];
#endif
      } else {
        *(float*)((char*)lds_b + off) = 0.0f;
      }
    }
  }
#if USE_ASYNC_LDS
  asm volatile("s_wait_asynccnt 0" ::: "memory");
#endif
  __syncthreads();

  // ---- per-wave 16x16 tile ----
  const int mt = bm * 8 + wv;
  if (mt * 16 < M) {
    const int   khalf = (lane >> 4) << 1;                // 0 or 2
    const float* aptr = A + (long)(mt * 16 + (lane & 15)) * KK;
    const char*  bptr = (const char*)lds_b + ((lane >> 4) * 128 + (lane & 15) * 8);

    v8f acc = {};
    for (int k0 = 0; k0 < KMAIN; k0 += 4) {              // fully in-bounds
      const int kk0 = k0 + khalf;
      v2f a, b;
      a[0] = aptr[kk0];
      a[1] = aptr[kk0 + 1];
      const float2 bb = *(const float2*)(bptr + (size_t)k0 * 64);
      b[0] = bb.x; b[1] = bb.y;
      acc = __builtin_amdgcn_wmma_f32_16x16x4_f32(
          false, a, false, b, (short)0, acc, false, false);
    }
    if (KPAD != KMAIN) {                                 // branchless masked tail
      const int kk0 = KMAIN + khalf;
      v2f a, b;
      float a0 = aptr[min(kk0, KK - 1)];
      float a1 = aptr[min(kk0 + 1, KK - 1)];
      a[0] = (kk0     < KK) ? a0 : 0.0f;
      a[1] = (kk0 + 1 < KK) ? a1 : 0.0f;
      const float2 bb = *(const float2*)(bptr + (size_t)KMAIN * 64);
      b[0] = bb.x; b[1] = bb.y;
      acc = __builtin_amdgcn_wmma_f32_16x16x4_f32(
          false, a, false, b, (short)0, acc, false, false);
    }

    const int bcol  = n0 + (lane & 15);
    const int mbase = mt * 16 + ((lane >> 4) << 3);
    if (bcol < N) {
      const float badd = bias ? bias[bcol] : 0.0f;
#pragma unroll
      for (int r = 0; r < 8; ++r) {
        float v = acc[r] + badd;
        if (do_relu) v = fmaxf(v, 0.0f);
        C[(long)(mbase + r) * N + bcol] = v;
      }
    }
  }
}

// ---------------- per-node attention coefficients (wave per (n,h)) ---------

__global__ void gat_attn_coef(const float* __restrict__ h,
                              const float* __restrict__ a_src,
                              const float* __restrict__ a_dst,
                              float* __restrict__ as, float* __restrict__ ad,
                              int N, int Hh, int F) {
  int wave = (int)((blockIdx.x * (long)blockDim.x + threadIdx.x) >> 5);
  int lane = threadIdx.x & 31;
  if (wave >= N * Hh) return;
  int n = wave / Hh, hh = wave - n * Hh;
  const float* hp = h + (long)n * Hh * F + (long)hh * F;
  const float* sp = a_src + hh * F;
  const float* dp = a_dst + hh * F;
  float s = 0.0f, d = 0.0f;
  for (int f = lane; f < F; f += 32) {
    float hv = hp[f];
    s += hv * sp[f];
    d += hv * dp[f];
  }
#pragma unroll
  for (int off = 16; off > 0; off >>= 1) {
    s += __shfl_down(s, off, 32);
    d += __shfl_down(d, off, 32);
  }
  if (lane == 0) { as[wave] = s; ad[wave] = d; }
}

// ---------------- edge pass 1: segment max of leaky-relu logits ------------

__global__ void gat_edge_max(const int* __restrict__ ei,
                             const float* __restrict__ as,
                             const float* __restrict__ ad,
                             float* __restrict__ emax, int Hh) {
  int e = blockIdx.x * blockDim.x + threadIdx.x;
  if (e >= ETOT) return;
  int s, d; edge_sd(ei, e, s, d);
  for (int h = 0; h < Hh; ++h) {
    float v = as[s * Hh + h] + ad[d * Hh + h];
    v = (v > 0.0f) ? v : 0.2f * v;
    atomicMaxF(&emax[d * Hh + h], v);
  }
}

// ---------------- edge pass 2: exp weights + segment sum -------------------

__global__ void gat_edge_expsum(const int* __restrict__ ei,
                                const float* __restrict__ as,
                                const float* __restrict__ ad,
                                const float* __restrict__ emax,
                                float* __restrict__ ee,
                                float* __restrict__ denom, int Hh) {
  int e = blockIdx.x * blockDim.x + threadIdx.x;
  if (e >= ETOT) return;
  int s, d; edge_sd(ei, e, s, d);
  for (int h = 0; h < Hh; ++h) {
    float v = as[s * Hh + h] + ad[d * Hh + h];
    v = (v > 0.0f) ? v : 0.2f * v;
    float w = __expf(v - emax[d * Hh + h]);
    ee[(long)e * Hh + h] = w;
    atomicAdd(&denom[d * Hh + h], w);
  }
}

// ---------------- edge pass 3: scatter weighted messages (wave per edge) ---

__global__ void gat_edge_scatter(const int* __restrict__ ei,
                                 const float* __restrict__ h,
                                 const float* __restrict__ ee,
                                 const float* __restrict__ denom,
                                 float* __restrict__ out, int Hh, int F) {
  int wave = (int)((blockIdx.x * (long)blockDim.x + threadIdx.x) >> 5);
  int lane = threadIdx.x & 31;
  if (wave >= ETOT) return;
  int s, d; edge_sd(ei, wave, s, d);
  for (int hh = 0; hh < Hh; ++hh) {
    float w = ee[(long)wave * Hh + hh] / (denom[d * Hh + hh] + 1e-16f);
    const float* hp = h + (long)s * Hh * F + (long)hh * F;
    float* op = out + (long)d * Hh * F + (long)hh * F;
    for (int f = lane; f < F; f += 32)
      atomicAdd(&op[f], w * hp[f]);
  }
}

// ---------------- layer-1 finalize: bias + ELU -----------------------------

__global__ void gat_bias_elu(float* __restrict__ o, const float* __restrict__ bias,
                             long total, int C) {
  long i = (long)blockIdx.x * blockDim.x + threadIdx.x;
  long stride = (long)gridDim.x * blockDim.x;
  for (; i < total; i += stride) {
    int c = (int)(i % C);
    float v = o[i] + bias[c];
    o[i] = (v > 0.0f) ? v : (__expf(v) - 1.0f);
  }
}

// ---------------- layer-2 finalize: bias + ReLU + per-graph max pool -------
// Post-ReLU values are >= 0 and the pool buffer is 0-initialized, so integer
// atomicMax on float bits reproduces segment_max + isfinite cleanup exactly.

__global__ void gat_bias_relu_pool(const float* __restrict__ o,
                                   const float* __restrict__ bias,
                                   const int* __restrict__ batch,
                                   float* __restrict__ g,
                                   long total, int C) {
  long i = (long)blockIdx.x * blockDim.x + threadIdx.x;
  long stride = (long)gridDim.x * blockDim.x;
  for (; i < total; i += stride) {
    int n = (int)(i / C);
    int c = (int)(i % C);
    float v = fmaxf(o[i] + bias[c], 0.0f);
    atomicMax((int*)&g[(long)batch[n] * C + c], __float_as_int(v));
  }
}

// ---------------------------------------------------------------------------

static inline int cdiv(long a, long b) { return (int)((a + b - 1) / b); }

extern "C" void kernel_launch(void* const* d_in, const int* in_sizes, int n_in,
                              void* d_out, int out_size, void* d_ws, size_t ws_size,
                              hipStream_t stream) {
  const float* x    = (const float*)d_in[0];
  const int*   ei   = (const int*)d_in[1];
  const int*   batch= (const int*)d_in[2];
  const float* W1   = (const float*)d_in[3];
  const float* a_s1 = (const float*)d_in[4];
  const float* a_d1 = (const float*)d_in[5];
  const float* b1   = (const float*)d_in[6];
  const float* W2   = (const float*)d_in[7];
  const float* a_s2 = (const float*)d_in[8];
  const float* a_d2 = (const float*)d_in[9];
  const float* b2   = (const float*)d_in[10];
  const float* Wg   = (const float*)d_in[11];
  const float* bg   = (const float*)d_in[12];
  float* out = (float*)d_out;
  float* ws  = (float*)d_ws;

  // workspace layout (floats); total ~33.56M floats (~134 MB)
  float* h1  = ws + 0;            // [NN, HF]  15,000,000
  float* o1  = ws + 15000000;     // [NN, HF]  15,000,000
  float* as1 = ws + 30000000;     // [NN, NH]     200,000
  float* ad1 = ws + 30200000;     // [NN, NH]     200,000
  float* em1 = ws + 30400000;     // [NN, NH]     200,000
  float* dn1 = ws + 30600000;     // [NN, NH]     200,000
  float* ee  = ws + 30800000;     // [ETOT, NH] 2,760,000 (layer2 reuses)
  // phase 2 reuses the h1 region (dead after layer-1 edge_scatter)
  float* h2  = ws + 0;            // [NN, D2]   2,560,000
  float* o2  = ws + 2560000;      // [NN, D2]   2,560,000
  float* g   = ws + 5120000;      // [NG, D2]      65,536
  float* as2 = as1; float* ad2 = ad1; float* em2 = em1; float* dn2 = dn1;

  const unsigned int NEG_INF = 0xFF800000u;

  // ---------- phase 1 init ----------
  gat_fill<<<2048, BLOCK, 0, stream>>>((unsigned*)em1, NEG_INF, (long)NN * NH);
  gat_fill<<<2048, BLOCK, 0, stream>>>((unsigned*)dn1, 0u, (long)NN * NH);
  gat_fill<<<4096, BLOCK, 0, stream>>>((unsigned*)o1, 0u, (long)NN * HF);

  // ---------- layer 1 ----------
  {
    int ntiles = cdiv(HF, 16), mblk = cdiv(cdiv(NN, 16), 8);
    gat_wmma_gemm<FIN><<<mblk * ntiles, BLOCK, 0, stream>>>(
        x, W1, h1, NN, HF, nullptr, 0);
  }
  {
    long waves = (long)NN * NH;
    gat_attn_coef<<<cdiv(waves * 32, BLOCK), BLOCK, 0, stream>>>(
        h1, a_s1, a_d1, as1, ad1, NN, NH, FIN);
  }
  gat_edge_max<<<cdiv(ETOT, BLOCK), BLOCK, 0, stream>>>(ei, as1, ad1, em1, NH);
  gat_edge_expsum<<<cdiv(ETOT, BLOCK), BLOCK, 0, stream>>>(ei, as1, ad1, em1, ee, dn1, NH);
  gat_edge_scatter<<<cdiv((long)ETOT * 32, BLOCK), BLOCK, 0, stream>>>(
      ei, h1, ee, dn1, o1, NH, FIN);
  gat_bias_elu<<<4096, BLOCK, 0, stream>>>(o1, b1, (long)NN * HF, HF);

  // ---------- phase 2 init ----------
  gat_fill<<<256, BLOCK, 0, stream>>>((unsigned*)em2, NEG_INF, (long)NN);
  gat_fill<<<256, BLOCK, 0, stream>>>((unsigned*)dn2, 0u, (long)NN);
  gat_fill<<<2048, BLOCK, 0, stream>>>((unsigned*)o2, 0u, (long)NN * D2);
  gat_fill<<<256, BLOCK, 0, stream>>>((unsigned*)g, 0u, (long)NG * D2);

  // ---------- layer 2 ----------
  {
    int ntiles = cdiv(D2, 16), mblk = cdiv(cdiv(NN, 16), 8);
    gat_wmma_gemm<HF><<<mblk * ntiles, BLOCK, 0, stream>>>(
        o1, W2, h2, NN, D2, nullptr, 0);
  }
  {
    long waves = (long)NN;  // single head
    gat_attn_coef<<<cdiv(waves * 32, BLOCK), BLOCK, 0, stream>>>(
        h2, a_s2, a_d2, as2, ad2, NN, 1, D2);
  }
  gat_edge_max<<<cdiv(ETOT, BLOCK), BLOCK, 0, stream>>>(ei, as2, ad2, em2, 1);
  gat_edge_expsum<<<cdiv(ETOT, BLOCK), BLOCK, 0, stream>>>(ei, as2, ad2, em2, ee, dn2, 1);
  gat_edge_scatter<<<cdiv((long)ETOT * 32, BLOCK), BLOCK, 0, stream>>>(
      ei, h2, ee, dn2, o2, 1, D2);
  gat_bias_relu_pool<<<2048, BLOCK, 0, stream>>>(o2, b2, batch, g, (long)NN * D2, D2);

  // ---------- classifier: out = relu(g @ Wg + bg) ----------
  {
    int ntiles = cdiv(D2, 16), mblk = cdiv(cdiv(NG, 16), 8);
    gat_wmma_gemm<D2><<<mblk * ntiles, BLOCK, 0, stream>>>(
        g, Wg, out, NG, D2, bg, 1);
  }
}